// try10Net_25855703122041
// MI455X (gfx1250) — compile-verified
//
#include <hip/hip_runtime.h>
#include <math.h>

typedef float v2f __attribute__((ext_vector_type(2)));
typedef float v8f __attribute__((ext_vector_type(8)));

#define N_NODES 588
#define K_DEG   32

// ---------------------------------------------------------------------------
// Fused EdgeConv x2. One workgroup; features and h1 live in LDS.
// EdgeConv: h_i = max_e[Theta x_s] + (Phi - Theta) x_i + tb + pb
// (max distributes over the per-node constant term).
// ---------------------------------------------------------------------------
__global__ __launch_bounds__(640) void econv2_kernel(
    const float* __restrict__ feat, const int* __restrict__ src,
    const float* __restrict__ t1w, const float* __restrict__ t1b,
    const float* __restrict__ p1w, const float* __restrict__ p1b,
    const float* __restrict__ t2w, const float* __restrict__ t2b,
    const float* __restrict__ p2w, const float* __restrict__ p2b,
    float* __restrict__ xout /* [588] */)
{
    __shared__ float sx[N_NODES * 3];
    __shared__ float h1[N_NODES * 3];
    const int tid = threadIdx.x;

    for (int i = tid; i < N_NODES * 3; i += blockDim.x) sx[i] = feat[i];
    __syncthreads();

    if (tid < N_NODES) {
        const float w00 = t1w[0], w01 = t1w[1], w02 = t1w[2];
        const float w10 = t1w[3], w11 = t1w[4], w12 = t1w[5];
        const float w20 = t1w[6], w21 = t1w[7], w22 = t1w[8];
        float m0 = -INFINITY, m1 = -INFINITY, m2 = -INFINITY;
        const int ebase = tid * K_DEG;
        for (int e = 0; e < K_DEG; ++e) {
            const int s = src[ebase + e];
            const float s0 = sx[s * 3 + 0], s1 = sx[s * 3 + 1], s2 = sx[s * 3 + 2];
            m0 = fmaxf(m0, fmaf(w00, s0, fmaf(w01, s1, w02 * s2)));
            m1 = fmaxf(m1, fmaf(w10, s0, fmaf(w11, s1, w12 * s2)));
            m2 = fmaxf(m2, fmaf(w20, s0, fmaf(w21, s1, w22 * s2)));
        }
        const float d0 = sx[tid * 3 + 0], d1 = sx[tid * 3 + 1], d2 = sx[tid * 3 + 2];
        h1[tid * 3 + 0] = m0 + (p1w[0] - w00) * d0 + (p1w[1] - w01) * d1 + (p1w[2] - w02) * d2 + t1b[0] + p1b[0];
        h1[tid * 3 + 1] = m1 + (p1w[3] - w10) * d0 + (p1w[4] - w11) * d1 + (p1w[5] - w12) * d2 + t1b[1] + p1b[1];
        h1[tid * 3 + 2] = m2 + (p1w[6] - w20) * d0 + (p1w[7] - w21) * d1 + (p1w[8] - w22) * d2 + t1b[2] + p1b[2];
    }
    __syncthreads();

    if (tid < N_NODES) {
        const float u0 = t2w[0], u1 = t2w[1], u2 = t2w[2];
        float m = -INFINITY;
        const int ebase = tid * K_DEG;
        for (int e = 0; e < K_DEG; ++e) {
            const int s = src[ebase + e];
            m = fmaxf(m, fmaf(u0, h1[s * 3 + 0], fmaf(u1, h1[s * 3 + 1], u2 * h1[s * 3 + 2])));
        }
        const float d0 = h1[tid * 3 + 0], d1 = h1[tid * 3 + 1], d2 = h1[tid * 3 + 2];
        xout[tid] = m + (p2w[0] - u0) * d0 + (p2w[1] - u1) * d1 + (p2w[2] - u2) * d2 + t2b[0] + p2b[0];
    }
}

// ---------------------------------------------------------------------------
// GEMV via V_WMMA_F32_16X16X4_F32.  y = act(W @ x + b), W is [M,K] row-major.
// One wave per 16-row tile; x occupies column 0 of B.
// Main K loop is unguarded (K rounded down to a multiple of 4) with aligned
// v2f (b64) loads; one guarded tail iteration covers K%4 (only fc2, K=10).
// All guards are value selects so EXEC stays all-ones around every WMMA.
// ---------------------------------------------------------------------------
__global__ __launch_bounds__(256) void gemv_wmma_kernel(
    const float* __restrict__ W, const float* __restrict__ bias,
    const float* __restrict__ x, float* __restrict__ y,
    int M, int K, int apply_lrelu)
{
    const int lane  = threadIdx.x & 31;
    const int wave  = threadIdx.x >> 5;
    const int tile  = blockIdx.x * 8 + wave;
    if (tile * 16 >= M) return;                 // wave-uniform exit, EXEC intact

    const int  mloc   = lane & 15;              // row within tile
    const int  khalf  = (lane >> 4) * 2;        // lanes 0-15: K=0,1 ; 16-31: K=2,3
    const bool colsel = (mloc == 0);            // lanes carrying B column 0
    int row = tile * 16 + mloc;
    if (row >= M) row = M - 1;                  // clamp; extra rows never stored
    const float* Wrow = W + (size_t)row * (size_t)K;

    const int Kmain = K & ~3;
    v8f c = {};
    for (int k0 = 0; k0 < Kmain; k0 += 4) {
        // k0 + khalf is even -> 8-byte aligned b64 loads
        const v2f a  = *(const v2f*)(Wrow + k0 + khalf);
        const v2f xv = *(const v2f*)(x    + k0 + khalf);
        v2f b;
        b.x = colsel ? xv.x : 0.0f;
        b.y = colsel ? xv.y : 0.0f;
        c = __builtin_amdgcn_wmma_f32_16x16x4_f32(
                /*neg_a=*/false, a, /*neg_b=*/false, b,
                /*c_mod=*/(short)0, c, /*reuse_a=*/false, /*reuse_b=*/false);
    }
    if (Kmain < K) {                            // tail: K%4 in {1,2,3}
        const int k  = Kmain + khalf;
        const int ka = (k     < K) ? k     : K - 1;
        const int kb = (k + 1 < K) ? k + 1 : K - 1;
        const float wa = Wrow[ka], wb = Wrow[kb];
        const float xa = x[ka],    xb = x[kb];
        v2f a, b;
        a.x = (k     < K) ? wa : 0.0f;
        a.y = (k + 1 < K) ? wb : 0.0f;
        b.x = (colsel && k     < K) ? xa : 0.0f;
        b.y = (colsel && k + 1 < K) ? xb : 0.0f;
        c = __builtin_amdgcn_wmma_f32_16x16x4_f32(
                false, a, false, b, (short)0, c, false, false);
    }

    // Column N=0 of D: lane 0 holds M=tile*16+0..7, lane 16 holds M=tile*16+8..15
    if (colsel) {
        const int mbase = tile * 16 + (lane >> 4) * 8;
#pragma unroll
        for (int r = 0; r < 8; ++r) {
            const int m = mbase + r;
            if (m < M) {
                float v = c[r] + bias[m];
                if (apply_lrelu) v = (v > 0.0f) ? v : 0.01f * v;
                y[m] = v;
            }
        }
    }
}

// ---------------------------------------------------------------------------
extern "C" void kernel_launch(void* const* d_in, const int* in_sizes, int n_in,
                              void* d_out, int out_size, void* d_ws, size_t ws_size,
                              hipStream_t stream)
{
    (void)in_sizes; (void)n_in; (void)out_size; (void)ws_size;
    const float* feat = (const float*)d_in[0];
    const int*   src  = (const int*)  d_in[1];
    // d_in[2] = dst (structurally e/32 for this setup; unused)
    const float* t1w = (const float*)d_in[3];  const float* t1b = (const float*)d_in[4];
    const float* p1w = (const float*)d_in[5];  const float* p1b = (const float*)d_in[6];
    const float* t2w = (const float*)d_in[7];  const float* t2b = (const float*)d_in[8];
    const float* p2w = (const float*)d_in[9];  const float* p2b = (const float*)d_in[10];
    const float* fcw[6] = { (const float*)d_in[11], (const float*)d_in[13],
                            (const float*)d_in[15], (const float*)d_in[17],
                            (const float*)d_in[19], (const float*)d_in[21] };
    const float* fcb[6] = { (const float*)d_in[12], (const float*)d_in[14],
                            (const float*)d_in[16], (const float*)d_in[18],
                            (const float*)d_in[20], (const float*)d_in[22] };

    float* ws = (float*)d_ws;
    float* x0 = ws + 0;      // 588
    float* a1 = ws + 1024;   // 10
    float* a2 = ws + 2048;   // 128
    float* a3 = ws + 3072;   // 256
    float* a4 = ws + 4096;   // 512
    float* a5 = ws + 6144;   // 1024

    econv2_kernel<<<1, 640, 0, stream>>>(feat, src, t1w, t1b, p1w, p1b,
                                         t2w, t2b, p2w, p2b, x0);

    struct Layer { const float* in; float* out; int M; int K; int act; };
    const Layer layers[6] = {
        { x0, a1,   10,  588, 1 },
        { a1, a2,  128,   10, 1 },
        { a2, a3,  256,  128, 1 },
        { a3, a4,  512,  256, 1 },
        { a4, a5, 1024,  512, 1 },
        { a5, (float*)d_out, 1764, 1024, 0 },
    };
    for (int i = 0; i < 6; ++i) {
        const int tiles  = (layers[i].M + 15) / 16;
        const int blocks = (tiles + 7) / 8;
        gemv_wmma_kernel<<<blocks, 256, 0, stream>>>(
            fcw[i], fcb[i], layers[i].in, layers[i].out,
            layers[i].M, layers[i].K, layers[i].act);
    }
}